// MeanSquaredError3D_74698071212389
// MI455X (gfx1250) — compile-verified
//
#include <hip/hip_runtime.h>

typedef __attribute__((ext_vector_type(2))) float v2f;
typedef __attribute__((ext_vector_type(8))) float v8f;

#define COL 14
#define NJ  23
#define MROWS 16  // M padded to 16 rows (rows 14,15 = 0) -> unconditional LDS reads

// One wave32 per (b, j) task; 8 waves (256 threads) per block.
__global__ void pose_loss_kernel(const float* __restrict__ o,
                                 const float* __restrict__ h,
                                 const float* __restrict__ t,
                                 const int* __restrict__ v,
                                 float* __restrict__ partial,
                                 int nTasks) {
  __shared__ float Msh[MROWS * COL];  // row-major M[r][c], rows 14..15 zero
  __shared__ float wsum[8];

  const int tid = threadIdx.x;

  // ---- Build the 14x14 gaussian matrix M in LDS (scipy 'reflect' boundary),
  //      zero-padded to 16 rows so no LDS access needs a bounds guard. ----
  if (tid < MROWS * COL) {
    const double e0 = 1.0;
    const double e1 = 0.6065306597126334;     // exp(-0.5)
    const double e2 = 0.1353352832366127;     // exp(-2)
    const double e3 = 0.011108996538242306;   // exp(-4.5)
    const double e4 = 0.00033546262790251185; // exp(-8)
    const double sumw = e0 + 2.0 * (e1 + e2 + e3 + e4);
    const double ed[9] = {e4, e3, e2, e1, e0, e1, e2, e3, e4};
    int r = tid / COL, c = tid % COL;
    float acc = 0.f;
    if (r < COL) {
#pragma unroll
      for (int d = 0; d < 9; ++d) {
        int i = r + d - 4;
        i = ((i % (2 * COL)) + 2 * COL) % (2 * COL);
        if (i >= COL) i = 2 * COL - 1 - i;
        if (i == c) acc += (float)(ed[d] / sumw);
      }
    }
    Msh[tid] = acc;
  }
  __syncthreads();

  const int warp = tid >> 5;
  const int lane = tid & 31;
  const int task = blockIdx.x * 8 + warp;
  float result = 0.f;

  if (task < nTasks) {
    const int b = task / NJ, j = task % NJ;
    const float* hp = h + (size_t)task * (COL * COL);
    const int hf  = lane >> 4;   // half-wave select
    const int row = lane & 15;   // matrix row for A operand

    // ---- Load H once, directly in the WMMA f32 16x16x4 A-matrix layout ----
    // lane<16 owns cols (4t+0, 4t+1) of its row, lane>=16 owns (4t+2, 4t+3).
    float a0[4], a1[4];
    float bv = -1.f;
    int bi = 0;
    float sh = 0.f, sh2 = 0.f;
#pragma unroll
    for (int tt = 0; tt < 4; ++tt) {
      int k = 4 * tt + 2 * hf;
      float x0 = 0.f, x1 = 0.f;
      if (row < COL && k < COL) {
        x0 = hp[row * COL + k];
        x1 = hp[row * COL + k + 1];
        int idx = row * COL + k;
        if (x0 > bv) { bv = x0; bi = idx; }
        if (x1 > bv) { bv = x1; bi = idx + 1; }
        sh += x0 + x1;
        sh2 = fmaf(x0, x0, fmaf(x1, x1, sh2));
      }
      a0[tt] = x0; a1[tt] = x1;
    }

    // ---- Wave reductions: argmax (first-max tie-break), sum, sum-of-squares ----
#pragma unroll
    for (int m = 16; m; m >>= 1) {
      float ov = __shfl_xor(bv, m, 32);
      int   oi = __shfl_xor(bi, m, 32);
      if (ov > bv || (ov == bv && oi < bi)) { bv = ov; bi = oi; }
      sh  += __shfl_xor(sh,  m, 32);
      sh2 += __shfl_xor(sh2, m, 32);
    }
    const int ymax = bi / COL, xmax = bi % COL;

    // ---- Target parameters ----
    const float* tp = t + (size_t)task * 3;
    const float tx = tp[0], ty = tp[1], tz = tp[2];
    const int v0 = v[(size_t)task * 2];
    const int xi = (int)truncf(tx * (float)COL);
    const int yi = (int)truncf(ty * (float)COL);
    const bool inr = (xi >= 0) && (xi <= COL - 1) && (yi >= 0) && (yi <= COL - 1);
    const bool maskb = (v0 == 1) && inr;
    const int xc = min(max(xi, 0), COL - 1);
    const int yc = min(max(yi, 0), COL - 1);

    // ---- gy/gx column statistics (sum, sumsq, min, max) via half-wave butterflies ----
    const int li = lane & 15;
    const int qsel = hf ? xc : yc;                    // low half: gy, high half: gx
    const float gvv = Msh[li * COL + qsel];           // rows 14,15 read as 0 (padded)
    float s = gvv, q = gvv * gvv;
    float mnv = (li < COL) ? gvv : 1e30f;             // branchless cndmask selects
    float mxv = (li < COL) ? gvv : -1e30f;
#pragma unroll
    for (int m = 8; m; m >>= 1) {
      s += __shfl_xor(s, m, 32);
      q += __shfl_xor(q, m, 32);
      mnv = fminf(mnv, __shfl_xor(mnv, m, 32));
      mxv = fmaxf(mxv, __shfl_xor(mxv, m, 32));
    }
    const float Sgy = __shfl(s, 0, 32),  Sgx = __shfl(s, 16, 32);
    const float Qgy = __shfl(q, 0, 32),  Qgx = __shfl(q, 16, 32);
    const float mny = __shfl(mnv, 0, 32), mnx = __shfl(mnv, 16, 32);
    const float mxy = __shfl(mxv, 0, 32), mxx = __shfl(mxv, 16, 32);
    const float mn = mny * mnx, mx = mxy * mxx;       // valid: all entries >= 0
    const float Af = maskb ? (1.0f / (mx - mn)) : 0.f; // tt = Af*g + Bf
    const float Bf = -mn * Af;

    // ---- gy^T * H * gx via 4 chained V_WMMA_F32_16X16X4_F32 ----
    // B[k][n] = gx[k] for all n, so every D column equals H @ gx (fp32 exact).
    // Padded Msh rows make every gx/gy fetch unconditional (gx[14],gx[15] = 0).
    v8f acc = {};
#pragma unroll
    for (int tt = 0; tt < 4; ++tt) {
      int k = 4 * tt + 2 * hf;                        // k in [0,14]
      float b0 = Msh[k * COL + xc];
      float b1 = Msh[(k + 1) * COL + xc];             // k+1 in [1,15]
      v2f av; av.x = a0[tt]; av.y = a1[tt];
      v2f bb; bb.x = b0;     bb.y = b1;
      acc = __builtin_amdgcn_wmma_f32_16x16x4_f32(
          false, av, false, bb, (short)0, acc, false, false);
    }
    float p = 0.f;
#pragma unroll
    for (int vv = 0; vv < 8; ++vv) {
      int rr = vv + 8 * hf;                           // D row = vv + 8*half, in [0,15]
      float gyv = Msh[rr * COL + yc];                 // rows 14,15 read as 0
      p = fmaf(gyv, acc[vv], p);
    }
    const float GHG = p + __shfl_xor(p, 16, 32);      // low-half rows + high-half rows

    // Sum(h - tt)^2 expanded: tt = Af*g + Bf, g = gy (x) gx (rank-1, non-negative)
    const float Sg = Sgy * Sgx;
    const float Qg = Qgy * Qgx;
    const float d1 = sh2 - 2.f * Af * GHG - 2.f * Bf * sh
                   + Af * Af * Qg + 2.f * Af * Bf * Sg
                   + (float)(COL * COL) * Bf * Bf;

    if (lane == 0) {
      const float scale = (float)(1.0 / 14.0);
      const size_t obase =
          ((size_t)b * (3 * NJ) + j) * (COL * COL) + (size_t)ymax * COL + xmax;
      const float px = o[obase] + (float)xmax * scale;
      const float py = o[obase + (size_t)NJ * (COL * COL)] + (float)ymax * scale;
      const float pz = o[obase + (size_t)(2 * NJ) * (COL * COL)];
      const float dx = px - tx, dy = py - ty, dz = pz - tz;
      result = d1 + dx * dx + dy * dy + dz * dz;
    }
  }

  if (lane == 0) wsum[warp] = result;
  __syncthreads();
  if (tid == 0) {
    float ssum = 0.f;
#pragma unroll
    for (int i = 0; i < 8; ++i) ssum += wsum[i];
    partial[blockIdx.x] = ssum;
  }
}

// Deterministic fixed-tree final reduction; out = total / NJ.
__global__ void reduce_kernel(const float* __restrict__ partial,
                              float* __restrict__ out, int n) {
  __shared__ float sm[256];
  float s = 0.f;
  for (int i = threadIdx.x; i < n; i += 256) s += partial[i];
  sm[threadIdx.x] = s;
  __syncthreads();
  for (int w = 128; w; w >>= 1) {
    if ((int)threadIdx.x < w) sm[threadIdx.x] += sm[threadIdx.x + w];
    __syncthreads();
  }
  if (threadIdx.x == 0) out[0] = sm[0] / (float)NJ;
}

extern "C" void kernel_launch(void* const* d_in, const int* in_sizes, int n_in,
                              void* d_out, int out_size, void* d_ws, size_t ws_size,
                              hipStream_t stream) {
  (void)n_in; (void)out_size; (void)ws_size;
  const float* o = (const float*)d_in[0];
  const float* h = (const float*)d_in[1];
  const float* t = (const float*)d_in[2];
  const int*   v = (const int*)d_in[3];

  const int B = in_sizes[1] / (NJ * COL * COL);
  const int nTasks = B * NJ;
  const int blocks = (nTasks + 7) / 8;

  float* partial = (float*)d_ws;
  pose_loss_kernel<<<blocks, 256, 0, stream>>>(o, h, t, v, partial, nTasks);
  reduce_kernel<<<1, 256, 0, stream>>>(partial, (float*)d_out, blocks);
}